// NormalizingFlow_52020643889526
// MI455X (gfx1250) — compile-verified
//
#include <hip/hip_runtime.h>

typedef float v2f __attribute__((ext_vector_type(2)));
typedef float v4f __attribute__((ext_vector_type(4)));
typedef float v8f __attribute__((ext_vector_type(8)));

#define NFLOWS 32

// ---------------------------------------------------------------------------
// Kernel 1: compose the 32 affine maps into one (M, c) and compute log_det.
//   x <- x @ W^T + b   ==(column form)==   x <- W x + b
//   M <- W_i * M ; c <- W_i * c + b_i ; logdet += log|det W_i|
// Trivial scalar work; one lane does it.
// ---------------------------------------------------------------------------
__global__ void flow_compose_kernel(const float* __restrict__ W,
                                    const float* __restrict__ B,
                                    float* __restrict__ ws,
                                    float* __restrict__ logdet_out) {
    if (threadIdx.x != 0 || blockIdx.x != 0) return;
    float m00 = 1.f, m01 = 0.f, m10 = 0.f, m11 = 1.f;
    float cx = 0.f, cy = 0.f, logdet = 0.f;
    for (int i = 0; i < NFLOWS; ++i) {
        const float w00 = W[4*i+0], w01 = W[4*i+1];
        const float w10 = W[4*i+2], w11 = W[4*i+3];
        const float bx  = B[2*i+0], by  = B[2*i+1];
        const float n00 = w00*m00 + w01*m10;
        const float n01 = w00*m01 + w01*m11;
        const float n10 = w10*m00 + w11*m10;
        const float n11 = w10*m01 + w11*m11;
        const float ncx = w00*cx + w01*cy + bx;
        const float ncy = w10*cx + w11*cy + by;
        m00 = n00; m01 = n01; m10 = n10; m11 = n11; cx = ncx; cy = ncy;
        logdet += logf(fabsf(w00*w11 - w01*w10));
    }
    ws[0] = m00; ws[1] = m01; ws[2] = m10; ws[3] = m11;
    ws[4] = cx;  ws[5] = cy;
    *logdet_out = logdet;
}

// ---------------------------------------------------------------------------
// Kernel 2: streaming affine transform of 4M points via V_WMMA_F32_16X16X4_F32.
// One WMMA per 32-point tile:
//   A (16x4): two diagonal 2x2 copies of M (rows 0-1 on K0-1, rows 2-3 on K2-3)
//   B (4x16): column n = point (p0+2n) in K0-1 and point (p0+2n+1) in K2-3
//   C       : bias broadcast into D rows 0-3
// Per the ISA VGPR layouts this means:
//   A: lane0=(M00,M01), lane1=(M10,M11), lane18/19 = same (second block)
//   B: every lane loads its own point as a b64 (lane n -> p0+2n, lane 16+n -> p0+2n+1)
//   D: lane n<16 holds (outx,outy) of both its points in d0..d3 -> one b128 store
// Memory-bound kernel: 64 MB total traffic -> ~2.75 us at 23.3 TB/s.
// ---------------------------------------------------------------------------
__global__ void flow_apply_wmma_kernel(const float* __restrict__ x,
                                       const float* __restrict__ ws,
                                       float* __restrict__ out,
                                       int npoints) {
    const int lane          = threadIdx.x & 31;
    const unsigned waveId   = (blockIdx.x * blockDim.x + threadIdx.x) >> 5;
    const unsigned numWaves = (gridDim.x * blockDim.x) >> 5;
    const int ntiles        = npoints >> 5;   // 32 points per WMMA tile

    const float m00 = ws[0], m01 = ws[1], m10 = ws[2], m11 = ws[3];
    const float cx  = ws[4], cy  = ws[5];

    // A operand: block-diagonal composed matrix.
    v2f a = {0.f, 0.f};
    if (lane == 0 || lane == 18) { a.x = m00; a.y = m01; }
    if (lane == 1 || lane == 19) { a.x = m10; a.y = m11; }

    // C operand: bias into D rows 0..3 (lanes 0..15 carry rows 0..7).
    v8f cacc = {};
    if (lane < 16) {
        cacc[0] = cx; cacc[1] = cy; cacc[2] = cx; cacc[3] = cy;
    }

    for (unsigned t = waveId; t < (unsigned)ntiles; t += numWaves) {
        const unsigned p0 = t << 5;  // first point of this 32-point tile
        const unsigned pidx = p0 + ((lane < 16) ? (unsigned)(2 * lane)
                                                : (unsigned)(2 * (lane - 16) + 1));
        const v2f bvec = ((const v2f*)x)[pidx];   // 8B load per lane, 256B/tile

        const v8f d = __builtin_amdgcn_wmma_f32_16x16x4_f32(
            /*neg_a=*/false, a, /*neg_b=*/false, bvec,
            /*c_mod=*/(short)0, cacc, /*reuse_a=*/false, /*reuse_b=*/false);

        if (lane < 16) {
            v4f r = { d[0], d[1], d[2], d[3] };       // points p0+2n, p0+2n+1
            ((v4f*)out)[(p0 >> 1) + (unsigned)lane] = r;  // contiguous b128
        }
    }

    // Tail (npoints % 32 != 0) — not hit for 4194304, kept for generality.
    const int tail = ntiles << 5;
    for (int i = tail + (int)(blockIdx.x * blockDim.x + threadIdx.x);
         i < npoints; i += (int)(gridDim.x * blockDim.x)) {
        const v2f p = ((const v2f*)x)[i];
        v2f r;
        r.x = m00 * p.x + m01 * p.y + cx;
        r.y = m10 * p.x + m11 * p.y + cy;
        ((v2f*)out)[i] = r;
    }
}

extern "C" void kernel_launch(void* const* d_in, const int* in_sizes, int n_in,
                              void* d_out, int out_size, void* d_ws, size_t ws_size,
                              hipStream_t stream) {
    const float* x  = (const float*)d_in[0];   // [N, 2] f32
    const float* W  = (const float*)d_in[1];   // [32, 2, 2] f32
    const float* B  = (const float*)d_in[2];   // [32, 2] f32
    float* out      = (float*)d_out;           // [N*2] x_out, then [1] log_det
    float* ws       = (float*)d_ws;            // 6 floats: M (row-major), c

    const int npoints = in_sizes[0] / 2;

    flow_compose_kernel<<<1, 32, 0, stream>>>(W, B, ws, out + (out_size - 1));

    // 1024 blocks x 256 threads = 8192 waves; 131072 tiles -> 16 tiles/wave
    // (wave-uniform trip count keeps EXEC all-1s at every WMMA).
    flow_apply_wmma_kernel<<<1024, 256, 0, stream>>>(x, ws, out, npoints);
}